// SparseMemory_41970420417619
// MI455X (gfx1250) — compile-verified
//
#include <hip/hip_runtime.h>
#include <hip/hip_bf16.h>
#include <math.h>
#include <stdint.h>

typedef float v2f __attribute__((ext_vector_type(2)));
typedef float v8f __attribute__((ext_vector_type(8)));

// Problem constants
#define BB 32
#define MM 65536
#define WW 32
#define RR 4
#define KK 16
#define CC 65
#define NIN 256

// Workspace layout (float elements)
#define OFF_RQ       ((size_t)0)          // 32*4*32 = 4096
#define OFF_WV       ((size_t)4096)       // 32*32   = 1024
#define OFF_IG       ((size_t)5120)       // 32*65   = 2080
#define OFF_WG       ((size_t)7200)       // 32
#define OFF_NEWROWS  ((size_t)7232)       // 32*65*32 = 66560
#define OFF_USAGE    ((size_t)73792)      // 32*65536 = 2097152
#define OFF_SCORES   ((size_t)2170944)    // 32*4*65536 = 8388608
#define OFF_LEAST    ((size_t)10559552)   // 32 ints
#define OFF_TOPK     ((size_t)10559584)   // 32*4*16 = 2048 ints

// ---------------- K1: linear projections ----------------
__global__ void k_proj(const float* __restrict__ xi,
                       const float* __restrict__ W_rq, const float* __restrict__ b_rq,
                       const float* __restrict__ W_wv, const float* __restrict__ b_wv,
                       const float* __restrict__ W_ig, const float* __restrict__ b_ig,
                       const float* __restrict__ W_wg, const float* __restrict__ b_wg,
                       float* __restrict__ rq, float* __restrict__ wv,
                       float* __restrict__ ig, float* __restrict__ wg) {
  __shared__ float xs[NIN];
  int b = blockIdx.x, t = threadIdx.x;
  xs[t] = xi[b * NIN + t];
  __syncthreads();
  const float* Wrow = nullptr;
  float bias = 0.f;
  int kind = -1, k = 0;
  if (t < 128)        { k = t;       Wrow = W_rq + (size_t)k * NIN; bias = b_rq[k]; kind = 0; }
  else if (t < 160)   { k = t - 128; Wrow = W_wv + (size_t)k * NIN; bias = b_wv[k]; kind = 1; }
  else if (t < 225)   { k = t - 160; Wrow = W_ig + (size_t)k * NIN; bias = b_ig[k]; kind = 2; }
  else if (t == 225)  { k = 0;       Wrow = W_wg;                   bias = b_wg[0]; kind = 3; }
  if (kind >= 0) {
    float acc = bias;
    for (int i = 0; i < NIN; ++i) acc += Wrow[i] * xs[i];
    if (kind == 0)      rq[b * 128 + k] = acc;
    else if (kind == 1) wv[b * 32 + k] = acc;
    else if (kind == 2) ig[b * CC + k] = 1.f / (1.f + expf(-acc));
    else                wg[b] = 1.f / (1.f + expf(-acc));
  }
}

// ---------------- K2: write phase + usage update + argmin ----------------
__global__ void k_write(const float* __restrict__ usage_in,
                        const float* __restrict__ rwts, const float* __restrict__ wwts,
                        const float* __restrict__ vis, const int* __restrict__ rpos,
                        const int* __restrict__ ts_ptr,
                        const float* __restrict__ ig, const float* __restrict__ wg,
                        const float* __restrict__ wv,
                        float* __restrict__ usage_ws, float* __restrict__ new_rows,
                        int* __restrict__ least_used) {
  __shared__ int posS[CC];
  __shared__ float rwgS[CC], wwgS[CC], reluS[CC], IS[CC], wwnS[CC];
  __shared__ float minuS;
  __shared__ float redv[256];
  __shared__ int redi[256];
  int b = blockIdx.x, t = threadIdx.x;
  int ts = *ts_ptr;
  for (int i = t; i < MM; i += 256)
    usage_ws[(size_t)b * MM + i] = usage_in[(size_t)b * MM + i];
  if (t < CC) {
    int p = rpos[b * CC + t];
    posS[t] = p;
    float rg = rwts[(size_t)b * MM + p];
    if (ts == 1) rg += 1.f;
    rwgS[t] = rg;
    wwgS[t] = wwts[(size_t)b * MM + p];
    reluS[t] = usage_in[(size_t)b * MM + p];
  }
  __syncthreads();
  if (t == 0) {
    float mn = reluS[0];
    for (int i = 1; i < CC; ++i) mn = fminf(mn, reluS[i]);
    minuS = mn;
  }
  __syncthreads();
  if (t < CC) {
    float u = (rwgS[t] + wwgS[t]) > 0.005f ? 1.f : 0.f;
    float I = (reluS[t] == minuS) ? 1.f : 0.f;
    IS[t] = I;
    reluS[t] = ((float)ts - reluS[t]) * u + reluS[t] * (1.f - u);
    float igv = ig[b * CC + t];
    wwnS[t] = wg[b] * (igv * rwgS[t] + (1.f - igv) * I);
  }
  __syncthreads();
  if (t == 0) {  // sequential scatter, last-wins (JAX .at[].set semantics)
    for (int i = 0; i < CC; ++i) usage_ws[(size_t)b * MM + posS[i]] = reluS[i];
  }
  __threadfence_block();
  __syncthreads();
  float bv = INFINITY; int bi = 0;
  for (int i = t; i < MM; i += 256) {
    float v = usage_ws[(size_t)b * MM + i];
    if (v < bv) { bv = v; bi = i; }
  }
  redv[t] = bv; redi[t] = bi;
  __syncthreads();
  for (int s = 128; s > 0; s >>= 1) {
    if (t < s) {
      float v2 = redv[t + s]; int i2 = redi[t + s];
      if (v2 < redv[t] || (v2 == redv[t] && i2 < redi[t])) { redv[t] = v2; redi[t] = i2; }
    }
    __syncthreads();
  }
  if (t == 0) least_used[b] = redi[0];
  for (int e = t; e < CC * WW; e += 256) {
    int i = e >> 5, w = e & 31;
    new_rows[(size_t)b * (CC * WW) + e] =
        vis[(size_t)b * (CC * WW) + e] * (1.f - IS[i]) + wwnS[i] * wv[b * 32 + w];
  }
}

// ---------------- K4: WMMA f32 score GEMM, async-LDS double-buffered ----------------
// Tile = 16 memory rows (2KB). DMA'd to LDS with 4x global_load_async_to_lds_b128
// (each moves 512 contiguous bytes), padded LDS stride 36 dwords -> conflict-free
// ds_load_b64 B-fragments. ASYNCcnt double buffering overlaps DMA with WMMA.
#define TPAD 36                      // padded row stride (floats)
#define TILE_F (16 * TPAD)           // floats per buffer
#define TILE_B (TILE_F * 4)          // bytes per buffer (2304)

__global__ void k_scores(const float* __restrict__ memory, const float* __restrict__ rq,
                         float* __restrict__ scores) {
  __shared__ float tile[8][2][TILE_F];   // 8 waves x 2 buffers = 36 KB LDS
  int slice = blockIdx.x;  // 0..31 -> 2048 rows each
  int b = blockIdx.y;      // batch
  int t = threadIdx.x;
  int wave = t >> 5, lane = t & 31;
  int m = lane & 15, hi = lane >> 4;

  // A fragments: rows 0..3 = queries, rest 0 (16x16x4 f32 layout)
  v2f a[8];
#pragma unroll
  for (int kb = 0; kb < 8; ++kb) {
    int k0 = kb * 4 + 2 * hi;
    float ax = 0.f, ay = 0.f;
    if (m < RR) {
      ax = rq[b * 128 + m * 32 + k0];
      ay = rq[b * 128 + m * 32 + k0 + 1];
    }
    a[kb].x = ax; a[kb].y = ay;
  }

  // per-lane DMA assignment: instruction i covers rows i*4+arow, segment aseg (16B)
  int arow = lane >> 3;    // 0..3
  int aseg = lane & 7;     // 0..7
  uint32_t gl_off  = (uint32_t)(arow * 128 + aseg * 16);  // within-tile global byte offset
  uint32_t lds_off = (uint32_t)(arow * (TPAD * 4) + aseg * 16);

  uint32_t ldsbase = (uint32_t)(uintptr_t)(&tile[wave][0][0]);
  uint32_t laddr[2][4];
#pragma unroll
  for (int bf = 0; bf < 2; ++bf)
#pragma unroll
    for (int i = 0; i < 4; ++i)
      laddr[bf][i] = ldsbase + (uint32_t)bf * TILE_B + (uint32_t)i * (4 * TPAD * 4) + lds_off;

  const char* mbytes = (const char*)(memory + (size_t)b * MM * WW);

  auto issue = [&](int p, int bf) {
    int rowbase = slice * 2048 + p * 128 + wave * 16;
    uint64_t g = (uint64_t)(uintptr_t)mbytes + (uint64_t)rowbase * 128u + gl_off;
    uint64_t g0 = g, g1 = g + 512, g2 = g + 1024, g3 = g + 1536;
    asm volatile("global_load_async_to_lds_b128 %0, %1, off" :: "v"(laddr[bf][0]), "v"(g0) : "memory");
    asm volatile("global_load_async_to_lds_b128 %0, %1, off" :: "v"(laddr[bf][1]), "v"(g1) : "memory");
    asm volatile("global_load_async_to_lds_b128 %0, %1, off" :: "v"(laddr[bf][2]), "v"(g2) : "memory");
    asm volatile("global_load_async_to_lds_b128 %0, %1, off" :: "v"(laddr[bf][3]), "v"(g3) : "memory");
  };

  issue(0, 0);
  for (int p = 0; p < 16; ++p) {
    int bf = p & 1;
    if (p + 1 < 16) {
      issue(p + 1, (p + 1) & 1);
      asm volatile("s_wait_asynccnt 0x4" ::: "memory");  // in-order: tile p has landed
    } else {
      asm volatile("s_wait_asynccnt 0x0" ::: "memory");
    }
    const float* bufp = &tile[wave][bf][0];
    v8f c = {};
#pragma unroll
    for (int kb = 0; kb < 8; ++kb) {
      v2f bv = *(const v2f*)(bufp + m * TPAD + kb * 4 + 2 * hi);  // conflict-free ds_load_b64
      c = __builtin_amdgcn_wmma_f32_16x16x4_f32(false, a[kb], false, bv,
                                                (short)0, c, false, false);
    }
    int rowbase = slice * 2048 + p * 128 + wave * 16;
    if (lane < 16) {
#pragma unroll
      for (int r = 0; r < RR; ++r)
        scores[(((size_t)(b * RR + r)) << 16) + rowbase + lane] = c[r];
    }
  }
}

// ---------------- K5: patch scores at scattered rows (last-wins) ----------------
__global__ void k_fixup(const int* __restrict__ rpos, const float* __restrict__ rq,
                        const float* __restrict__ new_rows, float* __restrict__ scores) {
  __shared__ int posS[CC];
  __shared__ int lastS[CC];
  int b = blockIdx.x, t = threadIdx.x;
  if (t < CC) posS[t] = rpos[b * CC + t];
  __syncthreads();
  if (t < CC) {
    int p = posS[t], last = 1;
    for (int j = t + 1; j < CC; ++j)
      if (posS[j] == p) { last = 0; break; }
    lastS[t] = last;
  }
  __syncthreads();
  for (int e = t; e < RR * CC; e += blockDim.x) {
    int r = e / CC, i = e % CC;
    if (lastS[i]) {
      float acc = 0.f;
      const float* qr = rq + b * 128 + r * 32;
      const float* nr = new_rows + (size_t)b * (CC * WW) + i * WW;
      for (int w = 0; w < WW; ++w) acc += qr[w] * nr[w];
      scores[(((size_t)(b * RR + r)) << 16) + posS[i]] = acc;
    }
  }
}

// ---------------- K6: exact top-16 per (b,r), tie-break lower index ----------------
__global__ void k_topk(const float* __restrict__ scores, int* __restrict__ topk) {
  __shared__ float sval[4096];
  __shared__ int sidx[4096];
  __shared__ float rv[256];
  __shared__ int ri[256];
  __shared__ int rs[256];
  int br = blockIdx.x;  // b*4 + r
  int t = threadIdx.x;
  const float* sp = scores + ((size_t)br << 16);
  float lv[16]; int li[16];
#pragma unroll
  for (int k = 0; k < 16; ++k) { lv[k] = -INFINITY; li[k] = 0x7FFFFFFF; }
  int worst = 0;
  for (int i = t; i < MM; i += 256) {
    float v = sp[i];
    float wv_ = lv[worst]; int wi_ = li[worst];
    if (v > wv_ || (v == wv_ && i < wi_)) {
      lv[worst] = v; li[worst] = i;
      worst = 0;
#pragma unroll
      for (int k = 1; k < 16; ++k)
        if (lv[k] < lv[worst] || (lv[k] == lv[worst] && li[k] > li[worst])) worst = k;
    }
  }
#pragma unroll
  for (int k = 0; k < 16; ++k) { sval[t * 16 + k] = lv[k]; sidx[t * 16 + k] = li[k]; }
  __syncthreads();
  for (int k = 0; k < 16; ++k) {
    float bv = -INFINITY; int bi = 0x7FFFFFFF; int bs = 0;
    for (int s = t; s < 4096; s += 256) {
      float v = sval[s]; int idx = sidx[s];
      if (v > bv || (v == bv && idx < bi)) { bv = v; bi = idx; bs = s; }
    }
    rv[t] = bv; ri[t] = bi; rs[t] = bs;
    __syncthreads();
    for (int s = 128; s > 0; s >>= 1) {
      if (t < s) {
        if (rv[t + s] > rv[t] || (rv[t + s] == rv[t] && ri[t + s] < ri[t])) {
          rv[t] = rv[t + s]; ri[t] = ri[t + s]; rs[t] = rs[t + s];
        }
      }
      __syncthreads();
    }
    if (t == 0) {
      topk[br * 16 + k] = ri[0];
      sval[rs[0]] = -INFINITY;
      sidx[rs[0]] = 0x7FFFFFFF;
    }
    __syncthreads();
  }
}

// ---------------- K7: gather + cosine softmax read ----------------
__global__ void k_read(const float* __restrict__ memory, const float* __restrict__ new_rows,
                       const float* __restrict__ rq, const int* __restrict__ rpos,
                       const int* __restrict__ topk, const int* __restrict__ least,
                       float* __restrict__ out) {
  __shared__ int posS[CC], rpS[CC], patchS[CC];
  __shared__ float vm[CC * WW];
  __shared__ float vnorm[CC], knorm[RR];
  __shared__ float sarr[RR * CC];
  int b = blockIdx.x, t = threadIdx.x;
  if (t < 64) posS[t] = topk[b * 64 + t];
  if (t == 64) posS[64] = least[b];
  if (t < CC) rpS[t] = rpos[b * CC + t];
  __syncthreads();
  if (t < CC) {
    int row = posS[t];
    row = row < 0 ? 0 : (row > MM - 1 ? MM - 1 : row);
    posS[t] = row;
  }
  __syncthreads();
  if (t < CC) {
    int row = posS[t], p = -1;
    for (int j = 0; j < CC; ++j)
      if (rpS[j] == row) p = j;  // keep LAST occurrence
    patchS[t] = p;
  }
  __syncthreads();
  for (int e = t; e < CC * WW; e += blockDim.x) {
    int c = e >> 5, w = e & 31;
    int p = patchS[c];
    vm[e] = (p >= 0) ? new_rows[(size_t)b * (CC * WW) + p * WW + w]
                     : memory[((size_t)b * MM + posS[c]) * WW + w];
  }
  __syncthreads();
  if (t < CC) {
    float s = 0.f;
    for (int w = 0; w < WW; ++w) { float x = vm[t * WW + w]; s += x * x; }
    vnorm[t] = sqrtf(s) + 1e-6f;
  }
  if (t < RR) {
    float s = 0.f;
    for (int w = 0; w < WW; ++w) { float x = rq[b * 128 + t * 32 + w]; s += x * x; }
    knorm[t] = sqrtf(s) + 1e-6f;
  }
  __syncthreads();
  for (int e = t; e < RR * CC; e += blockDim.x) {
    int r = e / CC, c = e % CC;
    float num = 0.f;
    for (int w = 0; w < WW; ++w) num += vm[c * WW + w] * rq[b * 128 + r * 32 + w];
    float den = (float)WW * vnorm[c] * knorm[r] + 1e-6f;
    sarr[e] = num / den;
  }
  __syncthreads();
  if (t < RR) {
    float mx = -INFINITY;
    for (int c = 0; c < CC; ++c) mx = fmaxf(mx, sarr[t * CC + c]);
    float sum = 0.f;
    for (int c = 0; c < CC; ++c) { float e_ = expf(sarr[t * CC + c] - mx); sarr[t * CC + c] = e_; sum += e_; }
    float inv = 1.f / sum;
    for (int c = 0; c < CC; ++c) sarr[t * CC + c] *= inv;
  }
  __syncthreads();
  for (int e = t; e < RR * WW; e += blockDim.x) {
    int r = e >> 5, w = e & 31;
    float acc = 0.f;
    for (int c = 0; c < CC; ++c) acc += sarr[r * CC + c] * vm[c * WW + w];
    out[b * 128 + e] = acc;
  }
}

extern "C" void kernel_launch(void* const* d_in, const int* in_sizes, int n_in,
                              void* d_out, int out_size, void* d_ws, size_t ws_size,
                              hipStream_t stream) {
  const float* xi   = (const float*)d_in[0];
  const float* mem  = (const float*)d_in[1];
  const float* rwts = (const float*)d_in[2];
  const float* wwts = (const float*)d_in[3];
  const float* usg  = (const float*)d_in[4];
  const float* vis  = (const float*)d_in[5];
  const float* W_rq = (const float*)d_in[6];
  const float* b_rq = (const float*)d_in[7];
  const float* W_wv = (const float*)d_in[8];
  const float* b_wv = (const float*)d_in[9];
  const float* W_ig = (const float*)d_in[10];
  const float* b_ig = (const float*)d_in[11];
  const float* W_wg = (const float*)d_in[12];
  const float* b_wg = (const float*)d_in[13];
  const int*   rpos = (const int*)d_in[14];
  const int*   ts   = (const int*)d_in[16];

  float* ws = (float*)d_ws;
  float* rq       = ws + OFF_RQ;
  float* wv       = ws + OFF_WV;
  float* ig       = ws + OFF_IG;
  float* wg       = ws + OFF_WG;
  float* newrows  = ws + OFF_NEWROWS;
  float* usage_ws = ws + OFF_USAGE;
  float* scores   = ws + OFF_SCORES;
  int*   least    = (int*)(ws + OFF_LEAST);
  int*   topk     = (int*)(ws + OFF_TOPK);

  k_proj<<<BB, 256, 0, stream>>>(xi, W_rq, b_rq, W_wv, b_wv, W_ig, b_ig, W_wg, b_wg,
                                 rq, wv, ig, wg);
  k_write<<<BB, 256, 0, stream>>>(usg, rwts, wwts, vis, rpos, ts, ig, wg, wv,
                                  usage_ws, newrows, least);
  k_scores<<<dim3(32, BB), 256, 0, stream>>>(mem, rq, scores);
  k_fixup<<<BB, 256, 0, stream>>>(rpos, rq, newrows, scores);
  k_topk<<<BB * RR, 256, 0, stream>>>(scores, topk);
  k_read<<<BB, 320, 0, stream>>>(mem, newrows, rq, rpos, topk, least, (float*)d_out);
}